// GNNClassifier_88648124990794
// MI455X (gfx1250) — compile-verified
//
#include <hip/hip_runtime.h>
#include <hip/hip_bf16.h>

// GNN SAGEConv pipeline for MI455X (gfx1250, wave32).
// Memory-bound workload (~2 GB traffic, ~80us floor @ 23.3 TB/s); GEMMs are
// tiny (3.3 GFLOP) and run on the WMMA pipe in exact fp32 via
// V_WMMA_F32_16X16X4_F32 so the fused node kernel keeps VALU free.

typedef __attribute__((ext_vector_type(2))) float v2f;
typedef __attribute__((ext_vector_type(4))) float v4f;
typedef __attribute__((ext_vector_type(8))) float v8f;

#define EMB 64
#define FSTRIDE 65   // LDS row stride: (row*65+k)%64 = (row+k)%64 -> conflict free

// ---------------------------------------------------------------- utilities
__global__ __launch_bounds__(256) void zero_f32(float* __restrict__ p, long long n) {
  long long i = (long long)blockIdx.x * blockDim.x + threadIdx.x;
  if (i < n) p[i] = 0.0f;
}

// x[n,c] = emb_table[tok[n], c]  (coalesced over c)
__global__ __launch_bounds__(256) void embed_kernel(const int* __restrict__ tok,
                                                    const float* __restrict__ emb,
                                                    float* __restrict__ x,
                                                    int n_nodes) {
  int tid = blockIdx.x * blockDim.x + threadIdx.x;
  if (tid >= n_nodes * EMB) return;
  int n = tid >> 6, c = tid & 63;
  x[tid] = emb[(size_t)tok[n] * EMB + c];
}

// cnt[dst[e]] += 1  (in-degree; identical for both layers, computed once)
__global__ __launch_bounds__(256) void degree_kernel(const int* __restrict__ dst,
                                                     float* __restrict__ cnt,
                                                     int n_edges) {
  int e = blockIdx.x * blockDim.x + threadIdx.x;
  if (e < n_edges) atomicAdd(&cnt[dst[e]], 1.0f);
}

// agg[dst[e], :] += f[src[e], :]   — 16 threads/edge, float4 per thread
__global__ __launch_bounds__(256) void scatter_kernel(const float* __restrict__ f,
                                                      const int* __restrict__ src,
                                                      const int* __restrict__ dst,
                                                      float* __restrict__ agg,
                                                      int n_edges) {
  int tid = blockIdx.x * blockDim.x + threadIdx.x;
  int e = tid >> 4;
  if (e >= n_edges) return;
  int c4 = (tid & 15) << 2;
  int s = src[e], d = dst[e];
  v4f v = *(const v4f*)(f + (size_t)s * EMB + c4);
  float* a = agg + (size_t)d * EMB + c4;
  atomicAdd(a + 0, v.x);
  atomicAdd(a + 1, v.y);
  atomicAdd(a + 2, v.z);
  atomicAdd(a + 3, v.w);
}

// ------------------------------------------------ fused SAGEConv node kernel
// out = relu( mean @ W_l + b_l + f @ W_r ),  mean = agg / max(cnt,1)
// Block = 128 threads = 4 waves; block tile = 64 nodes.
// Wave w owns output columns [16w, 16w+16); weight panels preloaded to VGPRs.
__global__ __launch_bounds__(128) void sage_wmma_kernel(const float* __restrict__ f,
                                                        const float* __restrict__ agg,
                                                        const float* __restrict__ cnt,
                                                        const float* __restrict__ Wl,
                                                        const float* __restrict__ bl,
                                                        const float* __restrict__ Wr,
                                                        float* __restrict__ out,
                                                        int n_nodes) {
  __shared__ float xs[64 * FSTRIDE];
  __shared__ float ms[64 * FSTRIDE];

  const int tid = threadIdx.x;
  const int node0 = blockIdx.x * 64;

  // Stage 64x64 feature + mean tiles into LDS (coalesced over columns).
  for (int idx = tid; idx < 64 * 64; idx += 128) {
    int r = idx >> 6, c = idx & 63;
    int node = node0 + r;
    float xv = 0.0f, mv = 0.0f;
    if (node < n_nodes) {
      xv = f[(size_t)node * EMB + c];
      float cc = cnt[node];
      mv = agg[(size_t)node * EMB + c] / fmaxf(cc, 1.0f);  // agg==0 when cnt==0
    }
    xs[r * FSTRIDE + c] = xv;
    ms[r * FSTRIDE + c] = mv;
  }
  __syncthreads();

  const int wave = tid >> 5;       // column tile (wave32)
  const int lane = tid & 31;
  const int m    = lane & 15;      // A: row M,  B: col N,  D: col N
  const int half = lane >> 4;      // K-half selector / D row offset
  const int col  = wave * 16 + m;  // global output column for this lane

  // Preload weight panels: B layout per 16x16x4_f32 -> VGPR0 holds K=k0+2*half.
  v2f wl[16], wr[16];
#pragma unroll
  for (int ks = 0; ks < 16; ++ks) {
    int k = ks * 4 + 2 * half;
    wl[ks] = v2f{Wl[(size_t)k * 64 + col], Wl[(size_t)(k + 1) * 64 + col]};
    wr[ks] = v2f{Wr[(size_t)k * 64 + col], Wr[(size_t)(k + 1) * 64 + col]};
  }
  const float bias = bl[col];

  for (int nt = 0; nt < 4; ++nt) {  // 16-node sub-tiles
    const float* xrow = xs + (nt * 16 + m) * FSTRIDE;
    const float* mrow = ms + (nt * 16 + m) * FSTRIDE;
    v8f acc = {bias, bias, bias, bias, bias, bias, bias, bias};
#pragma unroll
    for (int ks = 0; ks < 16; ++ks) {
      int k = ks * 4 + 2 * half;
      v2f am = {mrow[k], mrow[k + 1]};
      acc = __builtin_amdgcn_wmma_f32_16x16x4_f32(false, am, false, wl[ks],
                                                  (short)0, acc, false, false);
      v2f ax = {xrow[k], xrow[k + 1]};
      acc = __builtin_amdgcn_wmma_f32_16x16x4_f32(false, ax, false, wr[ks],
                                                  (short)0, acc, false, false);
    }
    // D layout: component r -> row (r + 8*half), lane%16 -> column.
#pragma unroll
    for (int r = 0; r < 8; ++r) {
      int row = node0 + nt * 16 + r + 8 * half;
      if (row < n_nodes) out[(size_t)row * EMB + col] = fmaxf(acc[r], 0.0f);
    }
  }
}

// ---------------------------------------------------------------- pooling
__global__ __launch_bounds__(256) void gcount_kernel(const int* __restrict__ batch,
                                                     float* __restrict__ gcnt,
                                                     int n_nodes) {
  int n = blockIdx.x * blockDim.x + threadIdx.x;
  if (n < n_nodes) atomicAdd(&gcnt[batch[n]], 1.0f);
}

__global__ __launch_bounds__(256) void pool_kernel(const float* __restrict__ h,
                                                   const int* __restrict__ batch,
                                                   float* __restrict__ pooled,
                                                   int n_nodes) {
  int tid = blockIdx.x * blockDim.x + threadIdx.x;
  if (tid >= n_nodes * EMB) return;
  int n = tid >> 6, c = tid & 63;
  atomicAdd(&pooled[(size_t)batch[n] * EMB + c], h[tid]);
}

// out[g,k] = b_out[k] + sum_c pooled[g,c]/max(gcnt,1) * W_out[c,k]
__global__ __launch_bounds__(256) void out_kernel(const float* __restrict__ pooled,
                                                  const float* __restrict__ gcnt,
                                                  const float* __restrict__ Wout,
                                                  const float* __restrict__ bout,
                                                  float* __restrict__ out,
                                                  int n_graphs, int n_cls) {
  int tid = blockIdx.x * blockDim.x + threadIdx.x;
  if (tid >= n_graphs * n_cls) return;
  int g = tid / n_cls, k = tid % n_cls;
  float c = gcnt[g];
  float inv = (c > 0.0f) ? 1.0f / c : 0.0f;
  float s = bout[k];
#pragma unroll
  for (int i = 0; i < EMB; ++i) s += pooled[(size_t)g * EMB + i] * inv * Wout[i * n_cls + k];
  out[tid] = s;
}

// ---------------------------------------------------------------- launcher
extern "C" void kernel_launch(void* const* d_in, const int* in_sizes, int n_in,
                              void* d_out, int out_size, void* d_ws, size_t ws_size,
                              hipStream_t stream) {
  const int*   tok   = (const int*)d_in[0];
  const int*   eidx  = (const int*)d_in[1];
  const int*   batch = (const int*)d_in[2];
  const float* emb   = (const float*)d_in[3];
  const float* Wl1   = (const float*)d_in[4];
  const float* bl1   = (const float*)d_in[5];
  const float* Wr1   = (const float*)d_in[6];
  const float* Wl2   = (const float*)d_in[7];
  const float* bl2   = (const float*)d_in[8];
  const float* Wr2   = (const float*)d_in[9];
  const float* Wout  = (const float*)d_in[10];
  const float* bout  = (const float*)d_in[11];
  float* out = (float*)d_out;

  const int N = in_sizes[0];        // 100000 nodes
  const int E = in_sizes[1] / 2;    // 1600000 edges
  const int NCLS = 2;
  const int G = out_size / NCLS;    // 512 graphs
  const int* src = eidx;
  const int* dst = eidx + E;

  // Workspace layout (floats): x | h | agg | cnt | pooled | gcnt
  float* ws     = (float*)d_ws;
  float* x      = ws;
  float* h      = x   + (size_t)N * EMB;
  float* agg    = h   + (size_t)N * EMB;
  float* cnt    = agg + (size_t)N * EMB;
  float* pooled = cnt + N;
  float* gcnt   = pooled + (size_t)G * EMB;

  // Zero the accumulated region (agg..gcnt are contiguous).
  long long z1 = (long long)N * EMB + N + (long long)G * EMB + G;
  zero_f32<<<(int)((z1 + 255) / 256), 256, 0, stream>>>(agg, z1);

  // Embedding gather.
  embed_kernel<<<(N * EMB + 255) / 256, 256, 0, stream>>>(tok, emb, x, N);

  // Degree (shared by both layers).
  degree_kernel<<<(E + 255) / 256, 256, 0, stream>>>(dst, cnt, E);

  // Layer 1: scatter messages, then fused mean+GEMM+relu.
  scatter_kernel<<<(E * 16 + 255) / 256, 256, 0, stream>>>(x, src, dst, agg, E);
  sage_wmma_kernel<<<(N + 63) / 64, 128, 0, stream>>>(x, agg, cnt, Wl1, bl1, Wr1, h, N);

  // Layer 2: re-zero agg, scatter h, fused conv writes back into x buffer.
  zero_f32<<<(int)(((long long)N * EMB + 255) / 256), 256, 0, stream>>>(agg, (long long)N * EMB);
  scatter_kernel<<<(E * 16 + 255) / 256, 256, 0, stream>>>(h, src, dst, agg, E);
  sage_wmma_kernel<<<(N + 63) / 64, 128, 0, stream>>>(h, agg, cnt, Wl2, bl2, Wr2, x, N);

  // Global mean pool + classifier head.
  gcount_kernel<<<(N + 255) / 256, 256, 0, stream>>>(batch, gcnt, N);
  pool_kernel<<<(N * EMB + 255) / 256, 256, 0, stream>>>(x, batch, pooled, N);
  out_kernel<<<(G * NCLS + 255) / 256, 256, 0, stream>>>(pooled, gcnt, Wout, bout, out, G, NCLS);
}